// Attention_msa_TwoStream_4664334483724
// MI455X (gfx1250) — compile-verified
//
#include <hip/hip_runtime.h>

// Problem constants (reference: B=1, N=3200, C=512, H=8, d=64)
#define NN   3200
#define CCH  512           // C
#define HH   8
#define DD   64
#define NT   (NN / 16)     // 200 16-row tiles
#define BLKP 10            // proposals per frame
#define EPSF 1e-8f
#define THRESH 0.75f

typedef __attribute__((ext_vector_type(2))) float v2f;
typedef __attribute__((ext_vector_type(8))) float v8f;

// ---------------------------------------------------------------------------
// Kernel 1: per-head L2 normalization of v_cls -> u, and write x_ori half of
// x_out (cols [C, 2C) of each row). One block per row n; wave w = head w.
// ---------------------------------------------------------------------------
__global__ __launch_bounds__(256) void prep_kernel(const float* __restrict__ vcls,
                                                   float* __restrict__ u,
                                                   float* __restrict__ xout) {
  const int n    = blockIdx.x;
  const int w    = threadIdx.x >> 5;   // head
  const int lane = threadIdx.x & 31;

  const float* row = vcls + (size_t)n * CCH + w * DD;
  v2f val = *(const v2f*)(row + lane * 2);
  float ss = val.x * val.x + val.y * val.y;
#pragma unroll
  for (int off = 16; off > 0; off >>= 1) ss += __shfl_xor(ss, off);
  const float sc = 1.0f / (sqrtf(ss) + EPSF);

  v2f o; o.x = val.x * sc; o.y = val.y * sc;
  *(v2f*)(u + (size_t)n * CCH + w * DD + lane * 2) = o;
  // x_ori: x_out[n][C + col] = v_cls[n][col]
  *(v2f*)(xout + (size_t)n * (2 * CCH) + CCH + w * DD + lane * 2) = val;
}

// ---------------------------------------------------------------------------
// Kernel 2: fused cosine-sim attention. Block = one 16-row n-tile, 8 waves
// (wave h = head h). Loops over 200 m-tiles:
//   S_h(16x16) = u_h[n-tile] @ u_h[m-tile]^T   (16x WMMA, 4 indep accum chains)
//   raw_mean tile = (1/8) sum_h S_h -> d_out sim region
//   x_h += (S_h * mask) @ v_h[m-tile]          (16x WMMA, 4 indep accumulators)
// Mask evaluated only for the ~2 m-tiles that can intersect the zero zone
// (uniform scalar branch -> EXEC stays all-ones for WMMA).
// ---------------------------------------------------------------------------
__global__ __launch_bounds__(256) void attn_kernel(const float* __restrict__ u,
                                                   const float* __restrict__ vcls,
                                                   float* __restrict__ rawmean,
                                                   float* __restrict__ xout) {
  __shared__ float sS[HH][16][16];   // per-wave S tile, 8 KB

  const int h    = threadIdx.x >> 5;
  const int lane = threadIdx.x & 31;
  const int li   = lane & 15;        // row/col within tile
  const int hi   = lane >> 4;        // lane-half selects K pair
  const int n0   = blockIdx.x * 16;

  // Per-lane mask constants (row n = n0 + li for A-side of GEMM-2)
  const int nrow = n0 + li;
  const int bs   = (nrow / BLKP) * BLKP;          // zero zone [bs, bs+9) except diag
  // Union of zero zones over the 16 rows of this n-tile:
  const int zlo  = (n0 / BLKP) * BLKP;
  const int zhi  = ((n0 + 15) / BLKP) * BLKP + (BLKP - 2);

  // A fragments (fixed n-tile): A[i][k] = u[n0+i][h*64+k]
  // ISA layout: lanes 0-15 -> K{0,1}, lanes 16-31 -> K{2,3} per 4-wide K step.
  v2f afrag[16];
  {
    const float* ar = u + (size_t)(n0 + li) * CCH + h * DD + 2 * hi;
#pragma unroll
    for (int ks = 0; ks < 16; ++ks) afrag[ks] = *(const v2f*)(ar + ks * 4);
  }

  v8f xacc[4];
#pragma unroll
  for (int c = 0; c < 4; ++c) xacc[c] = (v8f){};

  for (int mt = 0; mt < NT; ++mt) {
    const int m0 = mt * 16;

    // --- GEMM 1: S_h = A @ B,  B[k][j] = u[m0+j][h*64+k].
    // 4 independent accumulator chains to keep the matrix pipe busy.
    const float* br = u + (size_t)(m0 + li) * CCH + h * DD + 2 * hi;
    v8f s0 = (v8f){}, s1 = (v8f){}, s2 = (v8f){}, s3 = (v8f){};
#pragma unroll
    for (int ks = 0; ks < 16; ks += 4) {
      v2f b0 = *(const v2f*)(br + (ks + 0) * 4);
      v2f b1 = *(const v2f*)(br + (ks + 1) * 4);
      v2f b2 = *(const v2f*)(br + (ks + 2) * 4);
      v2f b3 = *(const v2f*)(br + (ks + 3) * 4);
      s0 = __builtin_amdgcn_wmma_f32_16x16x4_f32(false, afrag[ks + 0], false, b0,
                                                 (short)0, s0, false, false);
      s1 = __builtin_amdgcn_wmma_f32_16x16x4_f32(false, afrag[ks + 1], false, b1,
                                                 (short)0, s1, false, false);
      s2 = __builtin_amdgcn_wmma_f32_16x16x4_f32(false, afrag[ks + 2], false, b2,
                                                 (short)0, s2, false, false);
      s3 = __builtin_amdgcn_wmma_f32_16x16x4_f32(false, afrag[ks + 3], false, b3,
                                                 (short)0, s3, false, false);
    }
    s0 = s0 + s1;
    s2 = s2 + s3;
    s0 = s0 + s2;

    // D layout: VGPR r, lanes 0-15 -> M=r, lanes 16-31 -> M=r+8; N=li
#pragma unroll
    for (int r = 0; r < 8; ++r) sS[h][r + 8 * hi][li] = s0[r];
    __syncthreads();

    // raw_mean tile: 256 threads, one element each
    {
      const int i = threadIdx.x >> 4;
      const int j = threadIdx.x & 15;
      float acc = 0.0f;
#pragma unroll
      for (int hh = 0; hh < HH; ++hh) acc += sS[hh][i][j];
      rawmean[(size_t)(n0 + i) * NN + (m0 + j)] = acc * 0.125f;
    }

    // --- GEMM 2: x_h += (S_h * mask) @ v_h[m-tile]  (M=16, K=16, N=64)
    // A[i][k] = S_h[i][k] (masked), i = li, k = ks*4 + 2*hi + {0,1}
    v2f a2[4];
    if ((m0 <= zhi) && (m0 + 15 >= zlo)) {
      // Rare path: m-tile intersects this n-tile's zero zone.
#pragma unroll
      for (int ks = 0; ks < 4; ++ks) {
        const int k0 = ks * 4 + 2 * hi;
        const int mA = m0 + k0;
        const int mB = m0 + k0 + 1;
        const float kA = ((mA < bs) || (mA >= bs + BLKP - 1) || (mA == nrow)) ? 1.0f : 0.0f;
        const float kB = ((mB < bs) || (mB >= bs + BLKP - 1) || (mB == nrow)) ? 1.0f : 0.0f;
        a2[ks].x = sS[h][li][k0 + 0] * kA;
        a2[ks].y = sS[h][li][k0 + 1] * kB;
      }
    } else {
      // Common path (~197/200 tiles): no masking needed.
#pragma unroll
      for (int ks = 0; ks < 4; ++ks) {
        const int k0 = ks * 4 + 2 * hi;
        a2[ks].x = sS[h][li][k0 + 0];
        a2[ks].y = sS[h][li][k0 + 1];
      }
    }

    // ks-outer / c-inner: consecutive WMMAs hit independent accumulators.
#pragma unroll
    for (int ks = 0; ks < 4; ++ks) {
      const int k0 = ks * 4 + 2 * hi;
      const float* r0 = vcls + (size_t)(m0 + k0 + 0) * CCH + h * DD + li;
      const float* r1 = vcls + (size_t)(m0 + k0 + 1) * CCH + h * DD + li;
#pragma unroll
      for (int c = 0; c < 4; ++c) {
        v2f bf;
        bf.x = r0[c * 16];   // L2-resident
        bf.y = r1[c * 16];
        xacc[c] = __builtin_amdgcn_wmma_f32_16x16x4_f32(false, a2[ks], false, bf,
                                                        (short)0, xacc[c], false, false);
      }
    }
    __syncthreads();   // protect sS before next iteration overwrites
  }

  // Write x half of x_out: rows n0 + r + 8*hi, cols h*64 + c*16 + li
#pragma unroll
  for (int c = 0; c < 4; ++c)
#pragma unroll
    for (int r = 0; r < 8; ++r)
      xout[(size_t)(n0 + r + 8 * hi) * (2 * CCH) + h * DD + c * 16 + li] = xacc[c][r];
}

// ---------------------------------------------------------------------------
// Kernel 3: sim_round2, in place on the raw_mean region. One block per row.
//   v = raw * keep;  p = softmax(v);  out = (raw > 0.75 ? p : 0) / (sum + eps)
//   => out[m] = mask_m * exp(v_m - max) / (msum + eps*esum)
// ---------------------------------------------------------------------------
__global__ __launch_bounds__(256) void sim_kernel(float* __restrict__ sim) {
  const int n    = blockIdx.x;
  const int tid  = threadIdx.x;
  const int w    = tid >> 5;
  const int lane = tid & 31;
  __shared__ float redA[8], redB[8];

  float* row = sim + (size_t)n * NN;
  const int bs = (n / BLKP) * BLKP;   // zero zone [bs, bs+9) except diagonal

  float raw[13];                 // ceil(3200/256) = 13 per thread max
  float vmax = -3.0e38f;
  int idx = 0;
  for (int m = tid; m < NN; m += 256, ++idx) {
    const float r = row[m];
    raw[idx] = r;
    const float keep = ((m < bs) || (m >= bs + BLKP - 1) || (m == n)) ? 1.0f : 0.0f;
    vmax = fmaxf(vmax, r * keep);
  }
#pragma unroll
  for (int off = 16; off > 0; off >>= 1) vmax = fmaxf(vmax, __shfl_xor(vmax, off));
  if (lane == 0) redA[w] = vmax;
  __syncthreads();
  float gmax = redA[0];
#pragma unroll
  for (int i = 1; i < 8; ++i) gmax = fmaxf(gmax, redA[i]);
  __syncthreads();

  float esum = 0.0f, msum = 0.0f;
  idx = 0;
  for (int m = tid; m < NN; m += 256, ++idx) {
    const float keep = ((m < bs) || (m >= bs + BLKP - 1) || (m == n)) ? 1.0f : 0.0f;
    const float e = __expf(raw[idx] * keep - gmax);
    esum += e;
    if (raw[idx] > THRESH) msum += e;
  }
#pragma unroll
  for (int off = 16; off > 0; off >>= 1) {
    esum += __shfl_xor(esum, off);
    msum += __shfl_xor(msum, off);
  }
  if (lane == 0) { redA[w] = esum; redB[w] = msum; }
  __syncthreads();
  float ges = 0.0f, gms = 0.0f;
#pragma unroll
  for (int i = 0; i < 8; ++i) { ges += redA[i]; gms += redB[i]; }

  const float inv = 1.0f / (gms + EPSF * ges);
  idx = 0;
  for (int m = tid; m < NN; m += 256, ++idx) {
    const float keep = ((m < bs) || (m >= bs + BLKP - 1) || (m == n)) ? 1.0f : 0.0f;
    const float e = __expf(raw[idx] * keep - gmax);
    row[m] = (raw[idx] > THRESH) ? e * inv : 0.0f;
  }
}

// ---------------------------------------------------------------------------
// Launch. Inputs: [0]=x_cls (dead), [1]=v_cls, [2..7]=W/b (dead under DCE).
// d_out layout: x_out (3200*1024 floats) then sim_round2 (3200*3200 floats).
// d_ws: u (head-normalized v_cls), 3200*512 floats = 6.55 MB.
// ---------------------------------------------------------------------------
extern "C" void kernel_launch(void* const* d_in, const int* in_sizes, int n_in,
                              void* d_out, int out_size, void* d_ws, size_t ws_size,
                              hipStream_t stream) {
  (void)in_sizes; (void)n_in; (void)out_size; (void)ws_size;
  const float* vcls = (const float*)d_in[1];
  float* out = (float*)d_out;
  float* u   = (float*)d_ws;                     // NN*CCH floats
  float* sim = out + (size_t)NN * (2 * CCH);     // NN*NN floats

  prep_kernel<<<NN, 256, 0, stream>>>(vcls, u, out);
  attn_kernel<<<NT, 256, 0, stream>>>(u, vcls, sim, out);
  sim_kernel<<<NN, 256, 0, stream>>>(sim);
}